// StableMaxCrossEntropyLoss_35227321762365
// MI455X (gfx1250) — compile-verified
//
#include <hip/hip_runtime.h>
#include <hip/hip_bf16.h>

typedef float v2f __attribute__((ext_vector_type(2)));
typedef float v4f __attribute__((ext_vector_type(4)));
typedef float v8f __attribute__((ext_vector_type(8)));

#define BLK 256
#define UNROLL 4
#define EPSV 1e-12f

// stable_s(x) = x+1 if x>=0 else 1/(1-x)
// fast rcp + one Newton-Raphson refinement (~0.5 ulp)
__device__ __forceinline__ float stable_s(float x) {
    float d = 1.0f - x;
    float r = __builtin_amdgcn_rcpf(d);
    r = __builtin_fmaf(r, __builtin_fmaf(-d, r, 1.0f), r);
    return (x >= 0.0f) ? (x + 1.0f) : r;
}

__device__ __forceinline__ v2f s_pair(v4f x) {
    v2f a;
    a.x = stable_s(x.x) + stable_s(x.z);
    a.y = stable_s(x.y) + stable_s(x.w);
    return a;
}

// Pass 1: one block per row. Stream the row with NT b128 loads (4 independent
// loads in flight per wave), push stable_s sums through V_WMMA_F32_16X16X4_F32
// with B = ones (C accumulates row sums replicated over 16 columns), two
// independent C tiles to break the dependent-WMMA chain, then LDS tree-reduce.
__global__ __launch_bounds__(BLK)
void stablemax_denom_kernel(const float* __restrict__ logits,
                            float* __restrict__ denom,
                            int cols) {
    const int row = blockIdx.x;
    const int tid = threadIdx.x;
    const long long base = (long long)row * (long long)cols;
    const float* __restrict__ rp = logits + base;

    // Peel to 16B alignment (array base from hipMalloc is >=256B aligned).
    const int lead = (int)((4 - (base & 3LL)) & 3LL);
    const int n4 = (cols - lead) >> 2;            // aligned float4 count
    const int tail_start = lead + (n4 << 2);
    const int tail = cols - tail_start;           // 0..3

    const v4f* __restrict__ vp = (const v4f*)(rp + lead);

    v2f bmat; bmat.x = 1.0f; bmat.y = 1.0f;       // B = ones (4x16)
    v8f acc0 = {0.f, 0.f, 0.f, 0.f, 0.f, 0.f, 0.f, 0.f};
    v8f acc1 = acc0;

    // ---- Main loop: full tiles, branchless, 4 loads in flight per wave ----
    const int chunk = BLK * UNROLL;               // 1024 float4s / block-iter
    const int nfull = n4 / chunk;
    for (int j = 0; j < nfull; ++j) {
        const int i0 = j * chunk + tid;
        v4f x[UNROLL];
        #pragma unroll
        for (int u = 0; u < UNROLL; ++u)
            x[u] = __builtin_nontemporal_load(&vp[i0 + u * BLK]);
        #pragma unroll
        for (int u = 0; u < UNROLL; ++u) {
            v2f a = s_pair(x[u]);
            if ((u & 1) == 0)
                acc0 = __builtin_amdgcn_wmma_f32_16x16x4_f32(
                    false, a, false, bmat, (short)0, acc0, false, false);
            else
                acc1 = __builtin_amdgcn_wmma_f32_16x16x4_f32(
                    false, a, false, bmat, (short)0, acc1, false, false);
        }
    }

    // ---- Remainder: <= UNROLL masked iterations, block-uniform trip count ----
    const int rem_start = nfull * chunk;
    const int rem_iters = (n4 - rem_start + BLK - 1) / BLK;   // 0..UNROLL
    for (int j = 0; j < rem_iters; ++j) {
        const int i = rem_start + j * BLK + tid;
        v2f a; a.x = 0.0f; a.y = 0.0f;            // OOB lanes contribute 0
        if (i < n4) {
            v4f x = __builtin_nontemporal_load(&vp[i]);
            a = s_pair(x);
        }
        acc0 = __builtin_amdgcn_wmma_f32_16x16x4_f32(
            false, a, false, bmat, (short)0, acc0, false, false);
    }

    // Per-lane sum of both C tiles; cross-lane sum == 16 * wave_total.
    float t = (acc0[0] + acc1[0]) + (acc0[1] + acc1[1]) +
              (acc0[2] + acc1[2]) + (acc0[3] + acc1[3]) +
              (acc0[4] + acc1[4]) + (acc0[5] + acc1[5]) +
              (acc0[6] + acc1[6]) + (acc0[7] + acc1[7]);

    // Lead/tail scalars (<=6 total), scaled by 16 to match WMMA replication.
    float extra = 0.0f;
    if (tid < lead) extra += stable_s(rp[tid]);
    if (tid < tail) extra += stable_s(rp[tail_start + tid]);
    t += 16.0f * extra;

    __shared__ float red[BLK];
    red[tid] = t;
    __syncthreads();
    #pragma unroll
    for (int s = BLK / 2; s > 0; s >>= 1) {
        if (tid < s) red[tid] += red[tid + s];
        __syncthreads();
    }
    if (tid == 0) denom[row] = fmaxf(red[0] * (1.0f / 16.0f), EPSV);
}

// Pass 2: single block. Gather s(logits[r, target[r]]), compute
// loss_r = -log(max(s_t/denom_r, EPS)), deterministic mean.
__global__ __launch_bounds__(BLK)
void stablemax_loss_kernel(const float* __restrict__ logits,
                           const int* __restrict__ targets,
                           const float* __restrict__ denom,
                           float* __restrict__ out,
                           int rows, int cols) {
    const int tid = threadIdx.x;
    float accl = 0.0f;
    for (int r = tid; r < rows; r += BLK) {
        const int tgt = targets[r];
        const float st = stable_s(logits[(long long)r * (long long)cols + tgt]);
        float pt = st / denom[r];
        pt = fmaxf(pt, EPSV);
        accl += -logf(pt);
    }
    __shared__ float red[BLK];
    red[tid] = accl;
    __syncthreads();
    #pragma unroll
    for (int s = BLK / 2; s > 0; s >>= 1) {
        if (tid < s) red[tid] += red[tid + s];
        __syncthreads();
    }
    if (tid == 0) out[0] = red[0] / (float)rows;
}

extern "C" void kernel_launch(void* const* d_in, const int* in_sizes, int n_in,
                              void* d_out, int out_size, void* d_ws, size_t ws_size,
                              hipStream_t stream) {
    const float* logits = (const float*)d_in[0];
    const int* targets = (const int*)d_in[1];
    const int rows = in_sizes[1];                 // 4096
    const int cols = in_sizes[0] / rows;          // 50257
    float* denom = (float*)d_ws;                  // rows * 4 bytes of scratch

    stablemax_denom_kernel<<<rows, BLK, 0, stream>>>(logits, denom, cols);
    stablemax_loss_kernel<<<1, BLK, 0, stream>>>(logits, targets, denom,
                                                 (float*)d_out, rows, cols);
}